// SharedTTLinearLayer_87462714016022
// MI455X (gfx1250) — compile-verified
//
#include <hip/hip_runtime.h>

// Problem constants from the reference
#define BB   1024
#define NN   8
#define IN_  2048
#define OUT_ 2048
#define RR   16
#define KSPLIT 4
#define KCHUNK (IN_ / KSPLIT)   // 512

typedef __attribute__((ext_vector_type(2))) float v2f;
typedef __attribute__((ext_vector_type(8))) float v8f;

// ---------------------------------------------------------------------------
// Stage 1: build W1 in K-pair-packed layout.
//   W1p[n][q][p][2] = sum_r first[n,r] * middle[r, 2q+{0,1}, p]
// One thread per (n, q, p): 8*1024*16 = 131072 threads, float2 store.
// ---------------------------------------------------------------------------
__global__ void tt_build_w1p(const float* __restrict__ first,
                             const float* __restrict__ middle,
                             float* __restrict__ W1p) {
    int idx = blockIdx.x * blockDim.x + threadIdx.x;    // 0 .. 131071
    int p = idx & (RR - 1);
    int q = (idx >> 4) & (IN_ / 2 - 1);                 // K-pair index
    int n = idx >> 14;
    float a0 = 0.0f, a1 = 0.0f;
#pragma unroll
    for (int r = 0; r < RR; ++r) {
        float f = first[n * RR + r];
        a0 = fmaf(f, middle[((size_t)r * IN_ + 2 * q)     * RR + p], a0);
        a1 = fmaf(f, middle[((size_t)r * IN_ + 2 * q + 1) * RR + p], a1);
    }
    v2f w; w.x = a0; w.y = a1;
    *(v2f*)(W1p + ((size_t)idx << 1)) = w;   // ((n*1024 + q)*16 + p)*2
}

// ---------------------------------------------------------------------------
// Repack task core into K-pair layout: Tp[q][o][2] = task[2q+{0,1}][o]
// ---------------------------------------------------------------------------
__global__ void tt_pack_task(const float* __restrict__ task,
                             float* __restrict__ Tp) {
    int idx = blockIdx.x * blockDim.x + threadIdx.x;    // 0 .. 16383
    int o = idx & (OUT_ - 1);
    int q = idx >> 11;                                  // 0..7
    v2f t;
    t.x = task[(size_t)(2 * q)     * OUT_ + o];
    t.y = task[(size_t)(2 * q + 1) * OUT_ + o];
    *(v2f*)(Tp + ((size_t)idx << 1)) = t;
}

// ---------------------------------------------------------------------------
// Stage 2 (split-K): Zp[kc][b,n,p] = sum_{i in chunk kc} x[b,n,i]*W1[n,i,p]
// One wave per (n, 16-row batch tile, K-chunk): 2048 waves, 128 WMMA K-steps.
// A: float2 from x (row-major).  B: single float2 from packed W1p.
// ---------------------------------------------------------------------------
__global__ void tt_stage2(const float* __restrict__ x,
                          const float* __restrict__ W1p,
                          float* __restrict__ Zp) {
    const int lane = threadIdx.x & 31;
    const int job  = blockIdx.x * (blockDim.x >> 5) + (threadIdx.x >> 5); // 0..2047
    const int kc = job & (KSPLIT - 1);     // K-chunk
    const int mt = (job >> 2) & 63;        // batch tile
    const int n  = job >> 8;               // task index 0..7

    const int m0    = mt << 4;
    const int kbase = kc * KCHUNK;
    const int ncol  = lane & 15;
    const int koff  = (lane >> 4) << 1;    // 0 or 2

    const float* xrow = x + ((size_t)(m0 + ncol) * NN + n) * IN_ + kbase;
    const float* W1n  = W1p + (size_t)n * IN_ * RR;

    v8f c = {0.f, 0.f, 0.f, 0.f, 0.f, 0.f, 0.f, 0.f};
#pragma unroll 8
    for (int k0 = 0; k0 < KCHUNK; k0 += 4) {
        // A 16x4: lane -> row M=lane%16, K-pair {koff, koff+1}
        v2f a = *(const v2f*)(xrow + k0 + koff);
        // B 4x16 from packed W1p: pair index (kbase+k0+koff)/2, col ncol
        const int kk = kbase + k0 + koff;
        v2f b = *(const v2f*)(W1n + ((((size_t)kk >> 1) * RR + ncol) << 1));
        c = __builtin_amdgcn_wmma_f32_16x16x4_f32(
                false, a, false, b, (short)0, c, false, false);
    }

    // D layout: VGPR j -> M = j (lanes 0-15) or j+8 (lanes 16-31), N = lane%16
    float* Zc = Zp + (size_t)kc * (BB * NN * RR);
    const int rbase = m0 + ((lane >> 4) << 3);
#pragma unroll
    for (int j = 0; j < 8; ++j) {
        Zc[((size_t)(rbase + j) * NN + n) * RR + ncol] = c[j];
    }
}

// ---------------------------------------------------------------------------
// Stage 3: y[b,n,o] = sum_p (sum_kc Zp[kc][b,n,p]) * task[p,o]
// One wave per (n, m-tile, 256-wide output group). A tile summed over the 4
// K-split partials once, reused across 16 output tiles of 16 columns.
// ---------------------------------------------------------------------------
__global__ void tt_stage3(const float* __restrict__ Zp,
                          const float* __restrict__ Tp,
                          float* __restrict__ y) {
    const int lane = threadIdx.x & 31;
    const int job  = blockIdx.x * (blockDim.x >> 5) + (threadIdx.x >> 5); // 0..4095
    const int og = job & 7;          // output group: 256 columns each
    const int mt = (job >> 3) & 63;  // batch tile
    const int n  = job >> 9;         // task index 0..7

    const int m0    = mt << 4;
    const int obase = og << 8;
    const int ncol  = lane & 15;
    const int koff  = (lane >> 4) << 1;

    // A = Z tile [16 x 16] (4 chunks of 16x4), summed over split-K partials
    const size_t zoff = ((size_t)(m0 + ncol) * NN + n) * RR;
    v2f a[4];
#pragma unroll
    for (int kk = 0; kk < 4; ++kk) {
        v2f s = *(const v2f*)(Zp + zoff + kk * 4 + koff);
#pragma unroll
        for (int c = 1; c < KSPLIT; ++c) {
            s += *(const v2f*)(Zp + (size_t)c * (BB * NN * RR) + zoff + kk * 4 + koff);
        }
        a[kk] = s;
    }

    const int rbase = m0 + ((lane >> 4) << 3);
    for (int ot = 0; ot < 16; ++ot) {
        const int o0 = obase + (ot << 4);
        v8f acc = {0.f, 0.f, 0.f, 0.f, 0.f, 0.f, 0.f, 0.f};
#pragma unroll
        for (int kk = 0; kk < 4; ++kk) {
            const int krow = kk * 4 + koff;                 // even
            // B 4x16 from packed Tp: one float2 per lane
            v2f b = *(const v2f*)(Tp + ((((size_t)krow >> 1) * OUT_ + o0 + ncol) << 1));
            acc = __builtin_amdgcn_wmma_f32_16x16x4_f32(
                      false, a[kk], false, b, (short)0, acc, false, false);
        }
#pragma unroll
        for (int j = 0; j < 8; ++j) {
            y[((size_t)(rbase + j) * NN + n) * OUT_ + o0 + ncol] = acc[j];
        }
    }
}

// ---------------------------------------------------------------------------
extern "C" void kernel_launch(void* const* d_in, const int* in_sizes, int n_in,
                              void* d_out, int out_size, void* d_ws, size_t ws_size,
                              hipStream_t stream) {
    const float* x      = (const float*)d_in[0]; // [B, N, IN]
    const float* first  = (const float*)d_in[1]; // [N, R]
    const float* middle = (const float*)d_in[2]; // [R, IN, R]
    const float* task   = (const float*)d_in[3]; // [R, OUT]
    float* y = (float*)d_out;                    // [B, N, OUT]

    // Workspace layout (floats):
    //   W1p : N*IN*R                = 262144  (1 MB)   packed K-pairs
    //   Zp  : KSPLIT*B*N*R          = 524288  (2 MB)   split-K partials
    //   Tp  : R*OUT                 = 32768   (128 KB) packed K-pairs
    float* W1p = (float*)d_ws;
    float* Zp  = W1p + (size_t)NN * IN_ * RR;
    float* Tp  = Zp + (size_t)KSPLIT * BB * NN * RR;

    // Stage 1 + task repack (independent, tiny)
    tt_build_w1p<<<(NN * (IN_ / 2) * RR) / 256, 256, 0, stream>>>(first, middle, W1p);
    tt_pack_task<<<((RR / 2) * OUT_) / 256, 256, 0, stream>>>(task, Tp);

    // Stage 2: 2048 waves (8 per 256-thread block) -> 256 blocks
    tt_stage2<<<(NN * (BB / 16) * KSPLIT) / 8, 256, 0, stream>>>(x, W1p, Zp);

    // Stage 3: 4096 waves -> 512 blocks
    tt_stage3<<<(NN * (BB / 16) * (OUT_ / 256)) / 8, 256, 0, stream>>>(Zp, Tp, y);
}